// Net_12171937317449
// MI455X (gfx1250) — compile-verified
//
#include <hip/hip_runtime.h>

typedef __attribute__((ext_vector_type(2))) float v2f;
typedef __attribute__((ext_vector_type(8))) float v8f;

#define DIM 128
#define VEC_DIM 300
#define NE 32

// ---------------------------------------------------------------------------
// Leaf embedding: emb[n,:] = data_biases + relu(data_vecs[data[n]]) @ data_weights
// One block = 16 rows (nodes), 8 waves = 8 column tiles of 16.
// WMMA f32 16x16x4, K = 300.
// ---------------------------------------------------------------------------
__global__ __launch_bounds__(256)
void embed_kernel(const int* __restrict__ data,
                  const float* __restrict__ vecs,   // VOCAB x 300
                  const float* __restrict__ W,      // 300 x 128
                  const float* __restrict__ bias,   // 128
                  float* __restrict__ emb)          // N x 128
{
    __shared__ float lA[16][VEC_DIM + 4];           // padded stride 304 (even)
    const int tid  = threadIdx.x;
    const int row0 = blockIdx.x * 16;

    // stage relu(vecs[data[row]]) rows into LDS (shared by all 8 waves)
    for (int idx = tid; idx < 16 * VEC_DIM; idx += 256) {
        int r = idx / VEC_DIM, c = idx - r * VEC_DIM;
        int tok = data[row0 + r];
        float v = vecs[(long)tok * VEC_DIM + c];
        lA[r][c] = v > 0.f ? v : 0.f;
    }
    __syncthreads();

    const int lane = tid & 31;
    const int wave = tid >> 5;
    const int hf   = lane >> 4;            // 0: K pair {0,1}; 1: K pair {2,3}
    const int l16  = lane & 15;
    const int koff = hf * 2;
    const int col  = wave * 16 + l16;

    v8f c = {};
    for (int k = 0; k < VEC_DIM; k += 4) {
        v2f a = *(const v2f*)&lA[l16][k + koff];            // A[m=l16, k..k+1 | k+2..k+3]
        v2f b;
        b.x = W[(long)(k + koff)     * DIM + col];          // B[k, n]
        b.y = W[(long)(k + koff + 1) * DIM + col];
        c = __builtin_amdgcn_wmma_f32_16x16x4_f32(false, a, false, b,
                                                  (short)0, c, false, false);
    }

    float bc = bias[col];
#pragma unroll
    for (int r = 0; r < 8; ++r) {
        int m = r + hf * 8;
        emb[(long)(row0 + m) * DIM + col] = c[r] + bc;
    }
}

// ---------------------------------------------------------------------------
// Small helpers: zeroing, child counts, per-level edge-type binning.
// ---------------------------------------------------------------------------
__global__ void zero_f32(float* p, int n) {
    int i = blockIdx.x * blockDim.x + threadIdx.x;
    if (i < n) p[i] = 0.f;
}
__global__ void zero_i32(int* p, int n) {
    int i = blockIdx.x * blockDim.x + threadIdx.x;
    if (i < n) p[i] = 0;
}
__global__ void count_kernel(const int* __restrict__ parent, int* __restrict__ cnt,
                             int first_child, int n_children) {
    int i = blockIdx.x * blockDim.x + threadIdx.x;
    if (i < n_children) atomicAdd(&cnt[parent[first_child + i]], 1);
}
__global__ void hist_kernel(const int* __restrict__ edges, int cs, int nc,
                            int* __restrict__ hist) {
    int i = blockIdx.x * blockDim.x + threadIdx.x;
    if (i < nc) atomicAdd(&hist[edges[cs + i]], 1);
}
__global__ void scan32_kernel(const int* __restrict__ hist, int* __restrict__ bstart) {
    int e = threadIdx.x;                  // 32 threads
    int s = 0;
    for (int j = 0; j < e; ++j) s += hist[j];
    bstart[e] = s;
}
__global__ void scatter_kernel(const int* __restrict__ edges, int cs, int nc,
                               const int* __restrict__ bstart, int* __restrict__ cursor,
                               int* __restrict__ order) {
    int i = blockIdx.x * blockDim.x + threadIdx.x;
    if (i < nc) {
        int e = edges[cs + i];
        int p = atomicAdd(&cursor[e], 1);
        order[bstart[e] + p] = cs + i;    // global child index
    }
}

// ---------------------------------------------------------------------------
// Per-level batched GEMM: for one edge bucket, 16 children per tile:
//   contrib = relu(emb[child]) @ edge_weights[e] + edge_biases[e]
// atomically accumulated into agg[parent]. gridDim = (maxTiles, 32 buckets).
// ---------------------------------------------------------------------------
__global__ __launch_bounds__(256)
void level_gemm_kernel(const int* __restrict__ order,
                       const int* __restrict__ bstart,
                       const int* __restrict__ bcount,  // = hist
                       const int* __restrict__ parent,
                       const float* __restrict__ emb,
                       const float* __restrict__ eW,    // 32 x 128 x 128
                       const float* __restrict__ eB,    // 32 x 128
                       float* __restrict__ agg)         // NPAR x 128
{
    __shared__ float lA[16][DIM + 2];                   // stride 130 (even)
    __shared__ int   sChild[16];
    __shared__ int   sParent[16];

    const int e     = blockIdx.y;
    const int t     = blockIdx.x;
    const int cnt_e = bcount[e];
    if (t * 16 >= cnt_e) return;                        // block-uniform exit

    const int tid = threadIdx.x;
    if (tid < 16) {
        int g  = t * 16 + tid;
        int ch = (g < cnt_e) ? order[bstart[e] + g] : -1;
        sChild[tid]  = ch;
        sParent[tid] = (ch >= 0) ? parent[ch] : 0;
    }
    __syncthreads();

    for (int idx = tid; idx < 16 * DIM; idx += 256) {
        int r = idx >> 7, cdx = idx & 127;
        int ch = sChild[r];
        float v = 0.f;
        if (ch >= 0) {
            v = emb[(long)ch * DIM + cdx];
            v = v > 0.f ? v : 0.f;                      // relu(x)
        }
        lA[r][cdx] = v;                                 // zero rows for ragged tail
    }
    __syncthreads();

    const int lane = tid & 31;
    const int wave = tid >> 5;
    const int hf   = lane >> 4;
    const int l16  = lane & 15;
    const int koff = hf * 2;
    const int col  = wave * 16 + l16;
    const float* W = eW + (long)e * DIM * DIM;

    v8f c = {};
#pragma unroll 4
    for (int k = 0; k < DIM; k += 4) {
        v2f a = *(const v2f*)&lA[l16][k + koff];
        v2f b;
        b.x = W[(k + koff)     * DIM + col];
        b.y = W[(k + koff + 1) * DIM + col];
        c = __builtin_amdgcn_wmma_f32_16x16x4_f32(false, a, false, b,
                                                  (short)0, c, false, false);
    }

    float bc = eB[e * DIM + col];
#pragma unroll
    for (int r = 0; r < 8; ++r) {
        int m = r + hf * 8;
        if (t * 16 + m < cnt_e) {
            atomicAdd(&agg[(long)sParent[m] * DIM + col], c[r] + bc);
        }
    }
}

// emb[p,:] = (emb[p,:] + agg[p,:]) / (cnt[p] + 1)   (emb[p] still == data_emb[p])
__global__ void finalize_kernel(float* __restrict__ emb, const float* __restrict__ agg,
                                const int* __restrict__ cnt, int ps, int np) {
    int i = blockIdx.x * blockDim.x + threadIdx.x;
    if (i < np * DIM) {
        int p = ps + (i >> 7);
        int k = i & 127;
        long off = (long)p * DIM + k;
        emb[off] = (emb[off] + agg[off]) / ((float)cnt[p] + 1.0f);
    }
}

// ---------------------------------------------------------------------------
// Scoring head: scores[r,e] = ((relu(emb[r]) @ W_e + b_e) . score_weights)
// One block per edge e; 4 waves = 4 row tiles of the 64 roots.
// ---------------------------------------------------------------------------
__global__ __launch_bounds__(128)
void score_kernel(const float* __restrict__ emb,
                  const float* __restrict__ eW,
                  const float* __restrict__ eB,
                  const float* __restrict__ sw,    // 128
                  float* __restrict__ out)         // 64 x 32
{
    __shared__ float lA[64][DIM + 2];
    const int e   = blockIdx.x;
    const int tid = threadIdx.x;

    for (int idx = tid; idx < 64 * DIM; idx += 128) {
        int r = idx >> 7, cdx = idx & 127;
        float v = emb[(long)r * DIM + cdx];
        lA[r][cdx] = v > 0.f ? v : 0.f;
    }
    __syncthreads();

    const int lane = tid & 31;
    const int wave = tid >> 5;                  // row tile (0..3)
    const int hf   = lane >> 4;
    const int l16  = lane & 15;
    const int koff = hf * 2;
    const int mrow = wave * 16;
    const float* W = eW + (long)e * DIM * DIM;

    float partial[8];
#pragma unroll
    for (int r = 0; r < 8; ++r) partial[r] = 0.f;

    for (int ct = 0; ct < 8; ++ct) {
        const int col = ct * 16 + l16;
        v8f c = {};
#pragma unroll 4
        for (int k = 0; k < DIM; k += 4) {
            v2f a = *(const v2f*)&lA[mrow + l16][k + koff];
            v2f b;
            b.x = W[(k + koff)     * DIM + col];
            b.y = W[(k + koff + 1) * DIM + col];
            c = __builtin_amdgcn_wmma_f32_16x16x4_f32(false, a, false, b,
                                                      (short)0, c, false, false);
        }
        float s  = sw[col];
        float bb = eB[e * DIM + col];
#pragma unroll
        for (int r = 0; r < 8; ++r) partial[r] += (c[r] + bb) * s;
    }

    // reduce the 16 column lanes within each wave half (wave32 shuffles)
#pragma unroll
    for (int r = 0; r < 8; ++r) {
        float v = partial[r];
        v += __shfl_xor(v, 1, 32);
        v += __shfl_xor(v, 2, 32);
        v += __shfl_xor(v, 4, 32);
        v += __shfl_xor(v, 8, 32);
        if (l16 == 0) {
            int m = mrow + r + hf * 8;
            out[m * NE + e] = v;
        }
    }
}

// ---------------------------------------------------------------------------
// Host orchestration
// ---------------------------------------------------------------------------
extern "C" void kernel_launch(void* const* d_in, const int* in_sizes, int n_in,
                              void* d_out, int out_size, void* d_ws, size_t ws_size,
                              hipStream_t stream) {
    static const int LS[8] = {64, 128, 256, 512, 1024, 2048, 4096, 8192};
    int OFF[9]; OFF[0] = 0;
    for (int i = 0; i < 8; ++i) OFF[i + 1] = OFF[i] + LS[i];
    const int N    = OFF[8];   // 16320
    const int NPAR = OFF[7];   // 8128 (all possible parents)

    const int*   data = (const int*)d_in[0];
    const int*   par  = (const int*)d_in[1];
    const int*   edg  = (const int*)d_in[2];
    const float* vecs = (const float*)d_in[3];
    const float* dW   = (const float*)d_in[4];
    const float* dB   = (const float*)d_in[5];
    const float* eW   = (const float*)d_in[6];
    const float* eB   = (const float*)d_in[7];
    const float* sw   = (const float*)d_in[8];
    float* out = (float*)d_out;

    char* ws = (char*)d_ws;
    float* emb = (float*)ws;  ws += (size_t)N * DIM * sizeof(float);
    float* agg = (float*)ws;  ws += (size_t)NPAR * DIM * sizeof(float);
    int* cnt   = (int*)ws;    ws += (size_t)NPAR * sizeof(int);
    int* order = (int*)ws;    ws += (size_t)8192 * sizeof(int);
    int* hist  = (int*)ws;    ws += 32 * sizeof(int);
    int* bsta  = (int*)ws;    ws += 32 * sizeof(int);
    int* curs  = (int*)ws;    ws += 32 * sizeof(int);
    (void)in_sizes; (void)n_in; (void)out_size; (void)ws_size;

    // fresh per-call state (ws is not re-poisoned between replays)
    zero_f32<<<(NPAR * DIM + 255) / 256, 256, 0, stream>>>(agg, NPAR * DIM);
    zero_i32<<<(NPAR + 255) / 256, 256, 0, stream>>>(cnt, NPAR);
    count_kernel<<<((N - 64) + 255) / 256, 256, 0, stream>>>(par, cnt, 64, N - 64);

    embed_kernel<<<N / 16, 256, 0, stream>>>(data, vecs, dW, dB, emb);

    for (int d = 6; d >= 0; --d) {
        int cs = OFF[d + 1], ce = OFF[d + 2], nc = ce - cs;
        int ps = OFF[d],     np = OFF[d + 1] - OFF[d];

        zero_i32<<<1, 96, 0, stream>>>(hist, 96);   // hist + bstart + cursor
        hist_kernel<<<(nc + 255) / 256, 256, 0, stream>>>(edg, cs, nc, hist);
        scan32_kernel<<<1, 32, 0, stream>>>(hist, bsta);
        scatter_kernel<<<(nc + 255) / 256, 256, 0, stream>>>(edg, cs, nc, bsta, curs, order);

        dim3 grid((nc + 15) / 16, NE);              // overprovisioned; idle buckets exit
        level_gemm_kernel<<<grid, 256, 0, stream>>>(order, bsta, hist, par,
                                                    emb, eW, eB, agg);
        finalize_kernel<<<(np * DIM + 255) / 256, 256, 0, stream>>>(emb, agg, cnt, ps, np);
    }

    score_kernel<<<NE, 128, 0, stream>>>(emb, eW, eB, sw, out);
}